// GNN_50139448213879
// MI455X (gfx1250) — compile-verified
//
#include <hip/hip_runtime.h>

// GraphSAGE 3-layer GNN for MI455X (gfx1250, wave32).
// Bottleneck is the edge gather/scatter (~2.5 GB, fits in 192MB L2), so the
// GEMMs use full-precision V_WMMA_F32_16X16X4_F32 (free vs. the memory
// roofline, exact fp32 parity with the reference).

typedef __attribute__((ext_vector_type(2))) float v2f;
typedef __attribute__((ext_vector_type(8))) float v8f;

#define GNN_DIN 128
#define GNN_DH  128

// ---------------------------------------------------------------- utilities
__global__ void gnn_fill_zero(float* __restrict__ p, long n) {
  long i = (long)blockIdx.x * blockDim.x + threadIdx.x;
  if (i < n) p[i] = 0.0f;
}

__global__ void gnn_degree(const int* __restrict__ dst, float* __restrict__ deg, int E) {
  int e = blockIdx.x * blockDim.x + threadIdx.x;
  if (e < E) atomicAdd(&deg[dst[e]], 1.0f);
}

__global__ void gnn_invdeg(float* __restrict__ deg, int N) {
  int i = blockIdx.x * blockDim.x + threadIdx.x;
  if (i < N) deg[i] = 1.0f / fmaxf(deg[i], 1.0f);
}

// Scatter-add of 128-wide feature rows: 32 threads per edge, 4 channels each.
// float4 gather from x[src], 4x global_atomic_add_f32 into agg[dst].
__global__ void gnn_scatter(const float* __restrict__ x, const int* __restrict__ src,
                            const int* __restrict__ dst, float* __restrict__ agg, int E) {
  long t = (long)blockIdx.x * blockDim.x + threadIdx.x;
  long e = t >> 5;
  if (e >= E) return;
  int g = (int)(t & 31) * 4;
  int s = src[e], d = dst[e];
  const float4 v = *(const float4*)(x + (long)s * GNN_DH + g);
  float* a = agg + (long)d * GNN_DH + g;
  atomicAdd(a + 0, v.x);
  atomicAdd(a + 1, v.y);
  atomicAdd(a + 2, v.z);
  atomicAdd(a + 3, v.w);
}

// ------------------------------------------------------------- WMMA GEMM
// out[M,Dout] = (agg * inv_deg) @ Wl^T + X @ Wr^T + bias   (W row-major [Dout,Din])
// One wave32 per 16x16 output tile. f32 WMMA layouts (CDNA5 ISA):
//   A 16x4:  lane m = lane&15; VGPR j holds K = 2*(lane>>4) + j
//   B 4x16:  column n = lane&15; VGPR j holds K = 2*(lane>>4) + j (symmetric to A)
//   C/D:     column n = lane&15; VGPR r holds row M = r + 8*(lane>>4)
// Epilogue also accumulates per-column sum / sum-of-squares for BatchNorm.
__global__ void gnn_sage_gemm_wmma(
    const float* __restrict__ agg, const float* __restrict__ inv_deg,
    const float* __restrict__ X,
    const float* __restrict__ Wl, const float* __restrict__ Wr,
    const float* __restrict__ bias,
    float* __restrict__ out, float* __restrict__ stats,  // stats: [Dout]=sum, [Dout..2*Dout)=sumsq, may be null
    int N, int Din, int Dout) {
  const int lane  = threadIdx.x;        // 0..31 (wave32)
  const int mtile = blockIdx.x;
  const int ntile = blockIdx.y;
  const int mloc  = lane & 15;          // A row / B,C column within tile
  const int kg    = lane >> 4;          // K-group (lanes 16-31 hold K=2,3)

  int row = mtile * 16 + mloc;
  const int rowc = row < N ? row : N - 1;   // clamp: EXEC must stay all-1 for WMMA
  const int col  = ntile * 16 + mloc;

  const float bcol = bias[col];
  v8f acc;
#pragma unroll
  for (int r = 0; r < 8; ++r) acc[r] = bcol;

  const float s = inv_deg[rowc];
  const float* __restrict__ arow = agg + (long)rowc * Din;
  const float* __restrict__ xrow = X   + (long)rowc * Din;
  const float* __restrict__ wl   = Wl  + (long)col  * Din;
  const float* __restrict__ wr   = Wr  + (long)col  * Din;

  // term 1: (agg * inv_deg) @ Wl^T
  for (int k0 = 0; k0 < Din; k0 += 4) {
    const int ka = k0 + 2 * kg;
    v2f a, b;
    a[0] = arow[ka] * s;  a[1] = arow[ka + 1] * s;
    b[0] = wl[ka];        b[1] = wl[ka + 1];
    acc = __builtin_amdgcn_wmma_f32_16x16x4_f32(false, a, false, b, (short)0, acc,
                                                false, false);
  }
  // term 2: X @ Wr^T
  for (int k0 = 0; k0 < Din; k0 += 4) {
    const int ka = k0 + 2 * kg;
    v2f a, b;
    a[0] = xrow[ka];  a[1] = xrow[ka + 1];
    b[0] = wr[ka];    b[1] = wr[ka + 1];
    acc = __builtin_amdgcn_wmma_f32_16x16x4_f32(false, a, false, b, (short)0, acc,
                                                false, false);
  }

  // epilogue: store tile + BatchNorm partial stats
  const int n     = mloc;
  const int mbase = mtile * 16 + kg * 8;
  float ls = 0.0f, lq = 0.0f;
#pragma unroll
  for (int r = 0; r < 8; ++r) {
    const int gm = mbase + r;
    const float v = acc[r];
    if (gm < N) {
      out[(long)gm * Dout + ntile * 16 + n] = v;
      ls += v;
      lq += v * v;
    }
  }
  if (stats) {
    atomicAdd(&stats[ntile * 16 + n], ls);
    atomicAdd(&stats[Dout + ntile * 16 + n], lq);
  }
}

// ------------------------------------------------------------ BN + ReLU
// training-mode BN with biased batch stats: var = E[h^2] - mean^2
__global__ void gnn_bn_relu(float* __restrict__ h, const float* __restrict__ stats,
                            const float* __restrict__ g, const float* __restrict__ be,
                            int N, int Dout) {
  long i = (long)blockIdx.x * blockDim.x + threadIdx.x;
  if (i >= (long)N * Dout) return;
  const int c = (int)(i & (Dout - 1));   // Dout is a power of two (128)
  const float invN = 1.0f / (float)N;
  const float mean = stats[c] * invN;
  const float var  = fmaxf(stats[Dout + c] * invN - mean * mean, 0.0f);
  const float inv  = rsqrtf(var + 1e-5f);
  const float v = (h[i] - mean) * inv * g[c] + be[c];
  h[i] = v > 0.0f ? v : 0.0f;
}

// ------------------------------------------------------------ softmax(P=16)
__global__ void gnn_softmax16(float* __restrict__ out, int N, int P) {
  int i = blockIdx.x * blockDim.x + threadIdx.x;
  if (i >= N) return;
  float* row = out + (long)i * P;
  float m = row[0];
  for (int j = 1; j < P; ++j) m = fmaxf(m, row[j]);
  float e[16];
  float ssum = 0.0f;
  for (int j = 0; j < P; ++j) { e[j] = expf(row[j] - m); ssum += e[j]; }
  const float inv = 1.0f / ssum;
  for (int j = 0; j < P; ++j) row[j] = e[j] * inv;
}

// ---------------------------------------------------------------- launcher
extern "C" void kernel_launch(void* const* d_in, const int* in_sizes, int n_in,
                              void* d_out, int out_size, void* d_ws, size_t ws_size,
                              hipStream_t stream) {
  const float* x    = (const float*)d_in[0];
  const int*   edge = (const int*)d_in[1];
  const float* W1l = (const float*)d_in[2];
  const float* W1r = (const float*)d_in[3];
  const float* b1  = (const float*)d_in[4];
  const float* g1  = (const float*)d_in[5];
  const float* be1 = (const float*)d_in[6];
  const float* W2l = (const float*)d_in[7];
  const float* W2r = (const float*)d_in[8];
  const float* b2  = (const float*)d_in[9];
  const float* g2  = (const float*)d_in[10];
  const float* be2 = (const float*)d_in[11];
  const float* W3l = (const float*)d_in[12];
  const float* W3r = (const float*)d_in[13];
  const float* b3  = (const float*)d_in[14];

  const int N = in_sizes[0] / GNN_DIN;
  const int E = in_sizes[1] / 2;
  const int P = in_sizes[14];
  const int* src = edge;
  const int* dst = edge + E;

  // workspace carve-out (~155 MB): deg[N] | agg[N*128] | h1[N*128] | h2[N*128] | stats[2*128]
  float* ws    = (float*)d_ws;
  float* deg   = ws;
  float* agg   = deg + N;
  float* h1    = agg + (long)N * GNN_DH;
  float* h2    = h1  + (long)N * GNN_DH;
  float* stats = h2  + (long)N * GNN_DH;

  const long ND = (long)N * GNN_DH;
  auto cdiv = [](long a, long b) { return (int)((a + b - 1) / b); };
  const int mtiles = cdiv(N, 16);

  // degree -> inv_deg (in place)
  gnn_fill_zero<<<cdiv(N, 256), 256, 0, stream>>>(deg, N);
  gnn_degree<<<cdiv(E, 256), 256, 0, stream>>>(dst, deg, E);
  gnn_invdeg<<<cdiv(N, 256), 256, 0, stream>>>(deg, N);

  // ---------------- layer 1: x -> h1 ----------------
  gnn_fill_zero<<<cdiv(ND, 256), 256, 0, stream>>>(agg, ND);
  gnn_scatter<<<cdiv((long)E * 32, 256), 256, 0, stream>>>(x, src, dst, agg, E);
  gnn_fill_zero<<<1, 256, 0, stream>>>(stats, 2 * GNN_DH);
  {
    dim3 grid(mtiles, GNN_DH / 16);
    gnn_sage_gemm_wmma<<<grid, 32, 0, stream>>>(agg, deg, x, W1l, W1r, b1, h1, stats,
                                                N, GNN_DIN, GNN_DH);
  }
  gnn_bn_relu<<<cdiv(ND, 256), 256, 0, stream>>>(h1, stats, g1, be1, N, GNN_DH);

  // ---------------- layer 2: h1 -> h2 ----------------
  gnn_fill_zero<<<cdiv(ND, 256), 256, 0, stream>>>(agg, ND);
  gnn_scatter<<<cdiv((long)E * 32, 256), 256, 0, stream>>>(h1, src, dst, agg, E);
  gnn_fill_zero<<<1, 256, 0, stream>>>(stats, 2 * GNN_DH);
  {
    dim3 grid(mtiles, GNN_DH / 16);
    gnn_sage_gemm_wmma<<<grid, 32, 0, stream>>>(agg, deg, h1, W2l, W2r, b2, h2, stats,
                                                N, GNN_DH, GNN_DH);
  }
  gnn_bn_relu<<<cdiv(ND, 256), 256, 0, stream>>>(h2, stats, g2, be2, N, GNN_DH);

  // ---------------- layer 3: h2 -> d_out (P wide), then softmax ----------------
  gnn_fill_zero<<<cdiv(ND, 256), 256, 0, stream>>>(agg, ND);
  gnn_scatter<<<cdiv((long)E * 32, 256), 256, 0, stream>>>(h2, src, dst, agg, E);
  {
    dim3 grid(mtiles, P / 16);
    gnn_sage_gemm_wmma<<<grid, 32, 0, stream>>>(agg, deg, h2, W3l, W3r, b3,
                                                (float*)d_out, nullptr,
                                                N, GNN_DH, P);
  }
  gnn_softmax16<<<cdiv(N, 256), 256, 0, stream>>>((float*)d_out, N, P);
}